// FnRGNN_5085241279118
// MI455X (gfx1250) — compile-verified
//
#include <hip/hip_runtime.h>
#include <hip/hip_bf16.h>

#define NN 50000
#define NE 800000
#define HD 128

typedef __attribute__((ext_vector_type(2))) float v2f;
typedef __attribute__((ext_vector_type(8))) float v8f;

__device__ __forceinline__ float wave_sum32(float v) {
#pragma unroll
    for (int off = 16; off > 0; off >>= 1)
        v += __shfl_xor(v, off, 32);
    return v;
}

__global__ void zerof_kernel(float* p, int n) {
    int i = blockIdx.x * blockDim.x + threadIdx.x;
    if (i < n) p[i] = 0.0f;
}

// One wave (32 lanes) per node: rn[i] = 1 / max(||x_i||, 1e-8)
__global__ void node_norms_kernel(const float* __restrict__ x, float* __restrict__ rn, int n) {
    int i = blockIdx.x * 8 + (threadIdx.x >> 5);
    if (i >= n) return;
    int lane = threadIdx.x & 31;
    float4 a = ((const float4*)(x + (size_t)i * HD))[lane];
    float s = a.x * a.x + a.y * a.y + a.z * a.z + a.w * a.w;
    s = wave_sum32(s);
    if (lane == 0) rn[i] = 1.0f / fmaxf(sqrtf(s), 1e-8f);
}

// One wave per edge: cosine sim * exp(-gamma*sen_diff), clamp, degree atomic
__global__ void edge_weights_kernel(const float* __restrict__ x,
                                    const int* __restrict__ ei,
                                    const int* __restrict__ sens,
                                    const float* __restrict__ rn,
                                    float* __restrict__ ew,
                                    float* __restrict__ deg, int e_cnt) {
    int e = blockIdx.x * 8 + (threadIdx.x >> 5);
    if (e >= e_cnt) return;
    int lane = threadIdx.x & 31;
    int s = ei[e];
    int d = ei[e_cnt + e];
    float4 a = ((const float4*)(x + (size_t)s * HD))[lane];
    float4 b = ((const float4*)(x + (size_t)d * HD))[lane];
    float dot = a.x * b.x + a.y * b.y + a.z * b.z + a.w * b.w;
    dot = wave_sum32(dot);
    if (lane == 0) {
        float sim = dot * rn[s] * rn[d];
        float f = (sens[s] != sens[d]) ? 0.36787944117144233f : 1.0f; // exp(-1)
        float w = fmaxf(sim * f, 1e-4f);
        ew[e] = w;
        unsafeAtomicAdd(&deg[d], w);   // global_atomic_add_f32
    }
}

// dinv[i] = (deg[i] + 1)^-1/2   (in place over deg)
__global__ void make_dinv_kernel(float* dv, int n) {
    int i = blockIdx.x * blockDim.x + threadIdx.x;
    if (i < n) dv[i] = rsqrtf(dv[i] + 1.0f);
}

// ew[e] *= dinv[src] * dinv[dst]   (in place -> per-edge norm)
__global__ void edge_norm_kernel(float* __restrict__ ew, const int* __restrict__ ei,
                                 const float* __restrict__ dv, int e_cnt) {
    int e = blockIdx.x * blockDim.x + threadIdx.x;
    if (e >= e_cnt) return;
    ew[e] *= dv[ei[e]] * dv[ei[e_cnt + e]];
}

// Block = 8 waves, computes 16 rows x 128 cols of xt = A @ W using
// V_WMMA_F32_16X16X4_F32 (full f32). Epilogue writes raw xt AND the
// initialized aggregation buffer out = xt*dinv^2 + bias (self-loop + bias).
// relu_in applies ReLU to A while staging into LDS (fuses layer-1 ReLU).
__global__ void __launch_bounds__(256) gemm_wmma_kernel(
        const float* A, const float* __restrict__ W,
        const float* __restrict__ bias, const float* __restrict__ dv,
        float* __restrict__ xt, float* outp, int relu_in) {
    __shared__ float Alds[16 * 132];                 // stride 132 -> conflict-free reads
    const int tid = threadIdx.x;
    const int r0 = blockIdx.x * 16;
    const float4* Ain = (const float4*)(A + (size_t)r0 * HD);
#pragma unroll
    for (int i = 0; i < 2; ++i) {
        int idx = tid + i * 256;                     // 0..511 (16 rows x 32 float4)
        int row = idx >> 5, c4 = idx & 31;
        float4 v = Ain[row * 32 + c4];
        if (relu_in) {
            v.x = fmaxf(v.x, 0.0f); v.y = fmaxf(v.y, 0.0f);
            v.z = fmaxf(v.z, 0.0f); v.w = fmaxf(v.w, 0.0f);
        }
        *(float4*)(&Alds[row * 132 + c4 * 4]) = v;
    }
    __syncthreads();

    const int wave = tid >> 5;
    const int lane = tid & 31;
    const int half = lane >> 4;      // ISA layout: lanes 16-31 hold K+2
    const int mn   = lane & 15;
    const int colbase = wave * 16;   // each wave owns a 16-col tile

    v8f acc = {};
#pragma unroll
    for (int kt = 0; kt < 32; ++kt) {
        const int k = kt * 4 + half * 2;
        v2f a, b;
        a.x = Alds[mn * 132 + k];
        a.y = Alds[mn * 132 + k + 1];
        b.x = W[k * HD + colbase + mn];
        b.y = W[(k + 1) * HD + colbase + mn];
        acc = __builtin_amdgcn_wmma_f32_16x16x4_f32(
            /*neg_a=*/false, a, /*neg_b=*/false, b,
            /*c_mod=*/(short)0, acc, /*reuse_a=*/false, /*reuse_b=*/false);
    }

    const float bv = bias[colbase + mn];
#pragma unroll
    for (int r = 0; r < 8; ++r) {                    // D: VGPR r -> row r + 8*half
        const int row = r0 + r + half * 8;
        const float v = acc[r];
        const float di = dv[row];
        xt[(size_t)row * HD + colbase + mn] = v;
        outp[(size_t)row * HD + colbase + mn] = v * di * di + bv;
    }
}

// One wave per edge: out[dst] += xt[src] * norm[e]
__global__ void scatter_add_kernel(const float* __restrict__ xt,
                                   const int* __restrict__ ei,
                                   const float* __restrict__ nrm,
                                   float* __restrict__ outp, int e_cnt) {
    int e = blockIdx.x * 8 + (threadIdx.x >> 5);
    if (e >= e_cnt) return;
    int lane = threadIdx.x & 31;
    int s = ei[e];
    int d = ei[e_cnt + e];
    float w = nrm[e];
    float4 v = ((const float4*)(xt + (size_t)s * HD))[lane];
    float* o = outp + (size_t)d * HD + lane * 4;
    unsafeAtomicAdd(o + 0, v.x * w);
    unsafeAtomicAdd(o + 1, v.y * w);
    unsafeAtomicAdd(o + 2, v.z * w);
    unsafeAtomicAdd(o + 3, v.w * w);
}

// One wave per node: in-place ReLU of h, then y = h . Wc + bc
__global__ void finalize_kernel(float* h, const float* __restrict__ Wc,
                                const float* __restrict__ bc, float* __restrict__ y, int n) {
    int i = blockIdx.x * 8 + (threadIdx.x >> 5);
    if (i >= n) return;
    int lane = threadIdx.x & 31;
    float4* hp = (float4*)(h + (size_t)i * HD);
    float4 v = hp[lane];
    v.x = fmaxf(v.x, 0.0f); v.y = fmaxf(v.y, 0.0f);
    v.z = fmaxf(v.z, 0.0f); v.w = fmaxf(v.w, 0.0f);
    hp[lane] = v;
    float4 w = ((const float4*)Wc)[lane];
    float dot = v.x * w.x + v.y * w.y + v.z * w.z + v.w * w.w;
    dot = wave_sum32(dot);
    if (lane == 0) y[i] = dot + bc[0];
}

extern "C" void kernel_launch(void* const* d_in, const int* in_sizes, int n_in,
                              void* d_out, int out_size, void* d_ws, size_t ws_size,
                              hipStream_t stream) {
    const float* x  = (const float*)d_in[0];
    const int*   ei = (const int*)d_in[1];       // [2,E] int32 (JAX default x64-off)
    const int*   sa = (const int*)d_in[2];
    const float* W1 = (const float*)d_in[3];
    const float* b1 = (const float*)d_in[4];
    const float* W2 = (const float*)d_in[5];
    const float* b2 = (const float*)d_in[6];
    const float* Wc = (const float*)d_in[7];
    const float* bc = (const float*)d_in[8];

    const int n = NN;
    const int e = NE;

    float* y  = (float*)d_out;          // [N]
    float* h  = (float*)d_out + n;      // [N,128] — doubles as layer output buffer

    float* ws   = (float*)d_ws;
    float* rn   = ws;                   // [N]
    float* dv   = ws + n;               // [N]  deg -> dinv in place
    float* ew   = ws + 2 * n;           // [E]  ew -> norm in place
    float* xt   = ws + 2 * n + e;       // [N,128]

    const int B = 256;

    zerof_kernel<<<(n + B - 1) / B, B, 0, stream>>>(dv, n);
    node_norms_kernel<<<(n + 7) / 8, B, 0, stream>>>(x, rn, n);
    edge_weights_kernel<<<(e + 7) / 8, B, 0, stream>>>(x, ei, sa, rn, ew, dv, e);
    make_dinv_kernel<<<(n + B - 1) / B, B, 0, stream>>>(dv, n);
    edge_norm_kernel<<<(e + B - 1) / B, B, 0, stream>>>(ew, ei, dv, e);

    // Layer 1: h = relu-deferred( agg + xt*dinv^2 + b1 ); h init by GEMM epilogue
    gemm_wmma_kernel<<<n / 16, B, 0, stream>>>(x, W1, b1, dv, xt, h, 0);
    scatter_add_kernel<<<(e + 7) / 8, B, 0, stream>>>(xt, ei, ew, h, e);

    // Layer 2: ReLU of layer-1 fused into the A-tile load
    gemm_wmma_kernel<<<n / 16, B, 0, stream>>>(h, W2, b2, dv, xt, h, 1);
    scatter_add_kernel<<<(e + 7) / 8, B, 0, stream>>>(xt, ei, ew, h, e);

    // Final ReLU (in place on h) + classifier head
    finalize_kernel<<<(n + 7) / 8, B, 0, stream>>>(h, Wc, bc, y, n);
}